// CosineSimCodebook_89550068122198
// MI455X (gfx1250) — compile-verified
//
#include <hip/hip_runtime.h>
#include <stdint.h>

typedef __attribute__((ext_vector_type(2))) float v2f;
typedef __attribute__((ext_vector_type(4))) float v4f;
typedef __attribute__((ext_vector_type(8))) float v8f;

#define M_TOT 32768   // b*n = 8*4096
#define N_TOT 8192    // codebook size c
#define K_TOT 512     // d
#define BM 64
#define BN 256
#define BK 32
#define LDSP 36       // padded LDS row stride (floats): 16B aligned, conflict-free
#define NCHUNK (K_TOT / BK)

// monotone float -> uint32 key (order-preserving for all finite floats)
__device__ __forceinline__ uint32_t f2key(float f) {
  uint32_t u = __float_as_uint(f);
  return u ^ ((u >> 31) ? 0xFFFFFFFFu : 0x80000000u);
}

// CDNA5 async global -> LDS copy (ASYNCcnt-tracked, no staging VGPRs)
__device__ __forceinline__ void async_ld128(uint32_t lds_off, const float* gaddr) {
  asm volatile("global_load_async_to_lds_b128 %0, %1, off"
               :: "v"(lds_off), "v"(gaddr) : "memory");
}
__device__ __forceinline__ void wait_async0() {
  asm volatile("s_wait_asynccnt 0" ::: "memory");
}

// issue one K-chunk of A (2x b128/thread) + B (8x b128/thread) tile loads
__device__ __forceinline__ void issue_tile_loads(const float* aG, const float* bG,
                                                 int k0, uint32_t ldsA, uint32_t ldsB) {
#pragma unroll
  for (int i = 0; i < 2; ++i)
    async_ld128(ldsA + (uint32_t)i * (32u * LDSP * 4u),
                aG + (size_t)i * 32 * K_TOT + k0);
#pragma unroll
  for (int i = 0; i < 8; ++i)
    async_ld128(ldsB + (uint32_t)i * (32u * LDSP * 4u),
                bG + (size_t)i * 32 * K_TOT + k0);
}

__global__ void init_ws_kernel(unsigned long long* ws, int n) {
  int i = blockIdx.x * blockDim.x + threadIdx.x;
  if (i < n) ws[i] = 0ULL;
}

// GEMM dist = x @ emb^T with fused per-row argmax (packed u64 atomic-max merge).
// Double-buffered LDS, async global->LDS copies overlap the WMMA phase.
__global__ __launch_bounds__(256)
void cosine_gemm_argmax_kernel(const float* __restrict__ x,
                               const float* __restrict__ emb,
                               float* __restrict__ dist,
                               unsigned long long* __restrict__ ws) {
  __shared__ float As[2][BM * LDSP];
  __shared__ float Bs[2][BN * LDSP];

  const int t      = threadIdx.x;
  const int lane   = t & 31;
  const int wv     = t >> 5;       // 0..7
  const int wave_n = wv & 3;       // 4 waves across N
  const int wave_m = wv >> 2;      // 2 waves across M
  const int hf     = lane >> 4;    // half-wave selector (K pair / M+8)
  const int nn     = lane & 15;    // row-in-tile (A) / col-in-tile (B,D)

  const size_t m0 = (size_t)blockIdx.y * BM;
  const size_t n0 = (size_t)blockIdx.x * BN;

  v8f acc[2][4];
#pragma unroll
  for (int mt = 0; mt < 2; ++mt)
#pragma unroll
    for (int nt = 0; nt < 4; ++nt)
#pragma unroll
      for (int r = 0; r < 8; ++r) acc[mt][nt][r] = 0.0f;

  // ---- per-thread tile-load addressing (thread t covers row0, row0+32, ...) ----
  const int row0 = t >> 3;                       // 0..31
  const int quad = t & 7;                        // float4 slot in 32-float row
  const float* aG = x + (m0 + row0) * K_TOT + quad * 4;
  const float* bG = emb + (n0 + row0) * K_TOT + quad * 4;
  const uint32_t tileOff = (uint32_t)(row0 * LDSP + quad * 4) * 4u;
  uint32_t ldsA_cur = (uint32_t)(size_t)&As[0][0] + tileOff;
  uint32_t ldsA_nxt = (uint32_t)(size_t)&As[1][0] + tileOff;
  uint32_t ldsB_cur = (uint32_t)(size_t)&Bs[0][0] + tileOff;
  uint32_t ldsB_nxt = (uint32_t)(size_t)&Bs[1][0] + tileOff;

  // ---- fragment read pointers (ping-pong) ----
  const float* aFc = &As[0][(wave_m * 32 + nn) * LDSP + 2 * hf];
  const float* aFn = &As[1][(wave_m * 32 + nn) * LDSP + 2 * hf];
  const float* bFc = &Bs[0][(wave_n * 64 + nn) * LDSP + 2 * hf];
  const float* bFn = &Bs[1][(wave_n * 64 + nn) * LDSP + 2 * hf];

  // prologue: kick chunk 0 into current buffer
  issue_tile_loads(aG, bG, 0, ldsA_cur, ldsB_cur);

  for (int kc = 0; kc < NCHUNK; ++kc) {
    wait_async0();        // this wave's async writes for chunk kc have landed
    __syncthreads();      // all waves' data ready; prev compute on nxt buf done
    if (kc + 1 < NCHUNK)  // overlap next chunk's loads with this chunk's WMMAs
      issue_tile_loads(aG, bG, (kc + 1) * BK, ldsA_nxt, ldsB_nxt);

#pragma unroll
    for (int kk = 0; kk < BK; kk += 4) {
      v2f af[2], bf[4];
      af[0] = *(const v2f*)(aFc + kk);
      af[1] = *(const v2f*)(aFc + 16 * LDSP + kk);
      bf[0] = *(const v2f*)(bFc + kk);
      bf[1] = *(const v2f*)(bFc + 16 * LDSP + kk);
      bf[2] = *(const v2f*)(bFc + 32 * LDSP + kk);
      bf[3] = *(const v2f*)(bFc + 48 * LDSP + kk);
#pragma unroll
      for (int mt = 0; mt < 2; ++mt)
#pragma unroll
        for (int nt = 0; nt < 4; ++nt)
          acc[mt][nt] = __builtin_amdgcn_wmma_f32_16x16x4_f32(
              false, af[mt], false, bf[nt], (short)0, acc[mt][nt], false, false);
    }

    // ping-pong buffers
    { uint32_t tmp = ldsA_cur; ldsA_cur = ldsA_nxt; ldsA_nxt = tmp; }
    { uint32_t tmp = ldsB_cur; ldsB_cur = ldsB_nxt; ldsB_nxt = tmp; }
    { const float* tp = aFc; aFc = aFn; aFn = tp; }
    { const float* tp = bFc; bFc = bFn; bFn = tp; }
  }

  // ---- epilogue: NT-store dist, fused argmax ----
  const size_t mbase = m0 + (size_t)wave_m * 32;
  const size_t nbase = n0 + (size_t)wave_n * 64;
#pragma unroll
  for (int mt = 0; mt < 2; ++mt) {
    unsigned long long best[8];
#pragma unroll
    for (int r = 0; r < 8; ++r) best[r] = 0ULL;
#pragma unroll
    for (int nt = 0; nt < 4; ++nt) {
      union { v8f v; float f[8]; } u;
      u.v = acc[mt][nt];
      const size_t gn = nbase + nt * 16 + nn;
#pragma unroll
      for (int r = 0; r < 8; ++r) {
        const size_t gm = mbase + mt * 16 + r + 8 * hf;
        __builtin_nontemporal_store(u.f[r], &dist[gm * (size_t)N_TOT + gn]);
        // prefer smallest col on ties (jnp.argmax semantics): encode ~col
        unsigned long long p = ((unsigned long long)f2key(u.f[r]) << 32) |
                               (uint32_t)(0xFFFFFFFFu - (uint32_t)gn);
        best[r] = p > best[r] ? p : best[r];
      }
    }
#pragma unroll
    for (int r = 0; r < 8; ++r) {
      unsigned long long b = best[r];
#pragma unroll
      for (int msk = 1; msk <= 8; msk <<= 1) {   // stays within 16-lane half
        unsigned long long o = __shfl_xor(b, msk);
        b = o > b ? o : b;
      }
      if (nn == 0) {
        const size_t gm = mbase + mt * 16 + r + 8 * hf;
        atomicMax(&ws[gm], b);
      }
    }
  }
}

// decode winning index; gather codebook row; write embed_ind (as float)
__global__ __launch_bounds__(128)
void decode_gather_kernel(const unsigned long long* __restrict__ ws,
                          const float* __restrict__ emb,
                          float* __restrict__ q,
                          float* __restrict__ ind) {
  const int R = blockIdx.x;
  const unsigned long long p = ws[R];
  const uint32_t col = 0xFFFFFFFFu - (uint32_t)(p & 0xFFFFFFFFu);
  if (threadIdx.x == 0) ind[R] = (float)col;
  const float* src = emb + (size_t)col * K_TOT;
  float* dst = q + (size_t)R * K_TOT;
  v4f v = *(const v4f*)&src[threadIdx.x * 4];
  __builtin_nontemporal_store(v, (v4f*)&dst[threadIdx.x * 4]);
}

extern "C" void kernel_launch(void* const* d_in, const int* in_sizes, int n_in,
                              void* d_out, int out_size, void* d_ws, size_t ws_size,
                              hipStream_t stream) {
  const float* x   = (const float*)d_in[0];   // [8,4096,512]
  const float* emb = (const float*)d_in[1];   // [1,8192,512]
  float* out  = (float*)d_out;
  float* q    = out;                                   // [8,4096,512]
  float* ind  = out + (size_t)M_TOT * K_TOT;           // [8,4096]
  float* dist = ind + M_TOT;                           // [1,8,4096,8192]
  unsigned long long* ws = (unsigned long long*)d_ws;  // 32768 packed (key|~col)

  init_ws_kernel<<<(M_TOT + 255) / 256, 256, 0, stream>>>(ws, M_TOT);

  dim3 grid(N_TOT / BN, M_TOT / BM);   // 32 x 512 blocks
  cosine_gemm_argmax_kernel<<<grid, 256, 0, stream>>>(x, emb, dist, ws);

  decode_gather_kernel<<<M_TOT, 128, 0, stream>>>(ws, emb, q, ind);
}